// BatchNormSPD_32323923870228
// MI455X (gfx1250) — compile-verified
//
#include <hip/hip_runtime.h>

typedef float v2f __attribute__((ext_vector_type(2)));
typedef float v8f __attribute__((ext_vector_type(8)));

// ---------------------------------------------------------------------------
// Wave-collective helpers for 16x16 fp32 matrices resident in LDS (stride 16).
// CDNA5 wave32: one wave owns one matrix op. LDS (DS) ops are in-order per
// wave, so a wave_barrier (compiler scheduling fence) is enough for intra-wave
// cross-lane RAW through LDS. ALL helpers are __forceinline__ so that LDS
// pointers resolve to addrspace(3) after inlining -> ds_* ops, not flat_*.
// ---------------------------------------------------------------------------

__device__ __forceinline__ void wsync() {
  __builtin_amdgcn_wave_barrier();
  asm volatile("" ::: "memory");
}

// C = A * B  (16x16x16 f32) via 4 chained V_WMMA_F32_16X16X4_F32.
// A-layout (16x4 chunk): lane L holds A[L&15][k+2*(L>>4)], A[L&15][k+2*(L>>4)+1]
// B-layout (4x16 chunk): lane L holds B[k+2*(L>>4)][L&15], B[k+2*(L>>4)+1][L&15]
// D-layout: lane L vgpr v holds C[v + 8*(L>>4)][L&15]
__device__ __forceinline__ void w_mm(float* __restrict__ C,
                                     const float* A,
                                     const float* B) {
  const int lane = threadIdx.x & 31;
  const int col  = lane & 15;
  const int hi   = lane >> 4;
  const int row  = col;           // A-operand M index = lane & 15
  v8f acc = {};
#pragma unroll
  for (int kc = 0; kc < 4; ++kc) {
    const int k = kc * 4 + 2 * hi;
    v2f a, b;
    a.x = A[row * 16 + k];
    a.y = A[row * 16 + k + 1];
    b.x = B[k * 16 + col];
    b.y = B[(k + 1) * 16 + col];
    acc = __builtin_amdgcn_wmma_f32_16x16x4_f32(false, a, false, b,
                                                (short)0, acc, false, false);
  }
  wsync();
#pragma unroll
  for (int v = 0; v < 8; ++v) C[(v + 8 * hi) * 16 + col] = acc[v];
  wsync();
}

__device__ __forceinline__ void w_set_ident(float* M) {
  const int lane = threadIdx.x & 31;
#pragma unroll
  for (int i = 0; i < 8; ++i) {
    const int e = lane + 32 * i;
    M[e] = ((e >> 4) == (e & 15)) ? 1.0f : 0.0f;
  }
  wsync();
}

__device__ __forceinline__ void w_copy(float* D, const float* S) {
  const int lane = threadIdx.x & 31;
#pragma unroll
  for (int i = 0; i < 8; ++i) { const int e = lane + 32 * i; D[e] = S[e]; }
  wsync();
}

__device__ __forceinline__ void w_scale(float* D, const float* S, float a) {
  const int lane = threadIdx.x & 31;
#pragma unroll
  for (int i = 0; i < 8; ++i) { const int e = lane + 32 * i; D[e] = a * S[e]; }
  wsync();
}

__device__ __forceinline__ void w_axpy(float* D, const float* S, float a) {
  const int lane = threadIdx.x & 31;
#pragma unroll
  for (int i = 0; i < 8; ++i) { const int e = lane + 32 * i; D[e] += a * S[e]; }
  wsync();
}

__device__ __forceinline__ void w_zero(float* D) {
  const int lane = threadIdx.x & 31;
#pragma unroll
  for (int i = 0; i < 8; ++i) { const int e = lane + 32 * i; D[e] = 0.0f; }
  wsync();
}

// T := 1.5*I - 0.5*T   (Newton-Schulz update matrix)
__device__ __forceinline__ void w_ns_update(float* T) {
  const int lane = threadIdx.x & 31;
#pragma unroll
  for (int i = 0; i < 8; ++i) {
    const int e = lane + 32 * i;
    const float id = ((e >> 4) == (e & 15)) ? 1.5f : 0.0f;
    T[e] = id - 0.5f * T[e];
  }
  wsync();
}

__device__ __forceinline__ float w_fronorm(const float* S) {
  const int lane = threadIdx.x & 31;
  float s = 0.0f;
#pragma unroll
  for (int i = 0; i < 8; ++i) { const float v = S[lane + 32 * i]; s += v * v; }
#pragma unroll
  for (int off = 16; off >= 1; off >>= 1) s += __shfl_xor(s, off, 32);
  return sqrtf(s);
}

// Coupled Newton–Schulz: Y -> sqrt(A), Z -> A^{-1/2}.
// Frobenius pre-scaling (s >= lambda_max) guarantees convergence for SPD A.
__device__ __forceinline__ void w_ns_sqrt(const float* A, float* Y, float* Z,
                                          float* T1, float* T2, int iters) {
  float s = w_fronorm(A);
  s = fmaxf(s, 1e-20f);
  w_scale(Y, A, 1.0f / s);
  w_set_ident(Z);
  for (int it = 0; it < iters; ++it) {
    w_mm(T1, Z, Y);
    w_ns_update(T1);          // T1 = 1.5I - 0.5*Z*Y
    w_mm(T2, Y, T1); w_copy(Y, T2);
    w_mm(T2, T1, Z); w_copy(Z, T2);
  }
  const float sq = sqrtf(s);
  w_scale(Y, Y, sq);
  w_scale(Z, Z, 1.0f / sq);
}

// OUT = logm(A) for SPD A with eigenvalues moderately near 1.
// Inverse scaling-and-squaring: two NS square roots, then 9-term Gregory
// series for log(I+E), times 4. Clobbers A.
__device__ __forceinline__ void w_logm(float* A, float* OUT, float* Y, float* Z,
                                       float* T1, float* T2) {
  w_ns_sqrt(A, Y, Z, T1, T2, 14);
  w_copy(A, Y);
  w_ns_sqrt(A, Y, Z, T1, T2, 14);   // Y = A_orig^{1/4}
  // E = Y - I  (into Z)
  {
    const int lane = threadIdx.x & 31;
#pragma unroll
    for (int i = 0; i < 8; ++i) {
      const int e = lane + 32 * i;
      Z[e] = Y[e] - (((e >> 4) == (e & 15)) ? 1.0f : 0.0f);
    }
    wsync();
  }
  w_copy(T1, Z);     // P = E
  w_copy(OUT, Z);    // acc = E
  float sign = -1.0f;
  for (int k = 2; k <= 9; ++k) {
    w_mm(T2, T1, Z);
    w_copy(T1, T2);
    w_axpy(OUT, T1, sign / (float)k);
    sign = -sign;
  }
  w_scale(OUT, OUT, 4.0f);
}

// OUT = expm(A) via scaling (1/16) + 8-term Taylor + 4 squarings. Clobbers A.
__device__ __forceinline__ void w_expm(float* A, float* OUT, float* T1, float* T2) {
  w_scale(A, A, 1.0f / 16.0f);
  {
    const int lane = threadIdx.x & 31;
#pragma unroll
    for (int i = 0; i < 8; ++i) {
      const int e = lane + 32 * i;
      OUT[e] = A[e] + (((e >> 4) == (e & 15)) ? 1.0f : 0.0f);
    }
    wsync();
  }
  w_copy(T1, A);
  float invf = 1.0f;
  for (int k = 2; k <= 8; ++k) {
    w_mm(T2, T1, A); w_copy(T1, T2);
    invf /= (float)k;
    w_axpy(OUT, T1, invf);
  }
  for (int j = 0; j < 4; ++j) { w_mm(T2, OUT, OUT); w_copy(OUT, T2); }
}

// ---------------------------------------------------------------------------
// Kernel 1: partial sums for per-sample arithmetic means.
// Grid: 32 samples x 16 chunks = 512 blocks, 256 threads (one element each).
// ---------------------------------------------------------------------------
__global__ __launch_bounds__(256) void k_reduce(const float* __restrict__ X,
                                                float* __restrict__ part) {
  const int s = blockIdx.x >> 4;
  const int c = blockIdx.x & 15;
  const int e = threadIdx.x;
  const float* base = X + ((long)(s * 3200 + c * 200)) * 256 + e;
  float acc = 0.0f;
  for (int j = 0; j < 200; ++j) acc += base[(long)j * 256];
  part[(long)blockIdx.x * 256 + e] = acc;
}

// ---------------------------------------------------------------------------
// Kernel 2: all the Riemannian math. Single block, 8 waves.
// ---------------------------------------------------------------------------
__global__ __launch_bounds__(256) void k_math(const float* __restrict__ part,
                                              const float* __restrict__ running_mean,
                                              float* __restrict__ ws_xmean,
                                              float* __restrict__ ws_gim,
                                              float* __restrict__ out_rm) {
  __shared__ float WB[8][7][256];                 // per-wave workspace
  __shared__ float G[256], Gs[256], Gi[256], Lmean[256], MeanM[256];

  const int tid = threadIdx.x;
  const int wv  = tid >> 5;
  const int lane = tid & 31;

  // Phase 1: finish means: X_mean[s] (to ws) and grand mean G (to LDS).
  {
    float g = 0.0f;
    for (int s = 0; s < 32; ++s) {
      float acc = 0.0f;
      for (int c = 0; c < 16; ++c) acc += part[(long)(s * 16 + c) * 256 + tid];
      const float xm = acc * (1.0f / 3200.0f);
      ws_xmean[s * 256 + tid] = xm;
      g += xm;
    }
    G[tid] = g * (1.0f / 32.0f);
  }
  __threadfence();          // make ws_xmean visible across the block
  __syncthreads();

  // Phase 2: wave 0 computes Gs = G^{1/2}, Gi = G^{-1/2}.
  if (wv == 0) {
    w_copy(WB[0][0], G);
    w_ns_sqrt(WB[0][0], WB[0][1], WB[0][2], WB[0][3], WB[0][4], 14);
    w_copy(Gs, WB[0][1]);
    w_copy(Gi, WB[0][2]);
  }
  __syncthreads();

  // Phase 3: each wave computes logm(Gi * Xm_b * Gi) for b = wv, wv+8, ...
  {
    float* B0 = WB[wv][0]; float* B1 = WB[wv][1]; float* B2 = WB[wv][2];
    float* B3 = WB[wv][3]; float* B4 = WB[wv][4]; float* B5 = WB[wv][5];
    float* Lacc = WB[wv][6];
    w_zero(Lacc);
    for (int b = wv; b < 32; b += 8) {
#pragma unroll
      for (int i = 0; i < 8; ++i) {
        const int e = lane + 32 * i;
        B0[e] = ws_xmean[b * 256 + e];
      }
      wsync();
      w_mm(B1, Gi, B0);
      w_mm(B0, B1, Gi);                      // B0 = Gi Xm Gi
      w_logm(B0, B1, B2, B3, B4, B5);        // B1 = log-map
      w_axpy(Lacc, B1, 1.0f);
    }
  }
  __syncthreads();

  // Reduce log-maps to Lmean.
  {
    float acc = 0.0f;
    for (int w = 0; w < 8; ++w) acc += WB[w][6][tid];
    Lmean[tid] = acc * (1.0f / 32.0f);
  }
  __syncthreads();

  // Phase 4 (wave 0): Karcher mean, mean^{-1/2}, and the geodesic EMA.
  if (wv == 0) {
    float* B0 = WB[0][0]; float* B1 = WB[0][1]; float* B2 = WB[0][2];
    float* B3 = WB[0][3]; float* B4 = WB[0][4]; float* B5 = WB[0][5];
    float* B6 = WB[0][6];

    w_expm(Lmean, B0, B1, B2);               // B0 = exp(mean log-map)
    w_mm(B1, Gs, B0);
    w_mm(MeanM, B1, Gs);                     // MeanM = Karcher mean

    // ws_gim = MeanM^{-1/2}
    w_copy(B0, MeanM);
    w_ns_sqrt(B0, B1, B2, B3, B4, 14);       // B2 = inverse sqrt
#pragma unroll
    for (int i = 0; i < 8; ++i) { const int e = lane + 32 * i; ws_gim[e] = B2[e]; }
    wsync();

    // geodesic(running_mean, MeanM, 0.1)
#pragma unroll
    for (int i = 0; i < 8; ++i) { const int e = lane + 32 * i; B0[e] = running_mean[e]; }
    wsync();
    w_ns_sqrt(B0, B1, B3, B5, B6, 14);       // B1 = As, B3 = Ai
    w_mm(B0, B3, MeanM);
    w_mm(B4, B0, B3);                        // B4 = Ai MeanM Ai
    w_logm(B4, B0, B2, B5, B6, G);           // B0 = log(C)  (G reused as scratch)
    w_scale(B0, B0, 0.1f);
    w_expm(B0, B2, B5, B6);                  // B2 = C^0.1
    w_mm(B5, B1, B2);
    w_mm(B6, B5, B1);                        // B6 = As C^0.1 As
#pragma unroll
    for (int i = 0; i < 8; ++i) { const int e = lane + 32 * i; out_rm[e] = B6[e]; }
  }
}

// ---------------------------------------------------------------------------
// Kernel 3: Y = Gi * X * Gi for all 102400 matrices. Bandwidth-bound; one
// matrix per wave per iteration, Gi register-resident in both WMMA layouts.
// Output is streamed with non-temporal stores (never re-read; keep L2 for X).
// ---------------------------------------------------------------------------
__global__ __launch_bounds__(256) void k_apply(const float* __restrict__ X,
                                               const float* __restrict__ gim,
                                               float* __restrict__ out,
                                               int nmat) {
  __shared__ float gi[256];
  __shared__ float xb[8][256];
  __shared__ float tb[8][17 * 16];   // stride 17: kills LDS bank conflicts on A-layout reload

  const int tid = threadIdx.x;
  gi[tid] = gim[tid];
  __syncthreads();

  const int lane = tid & 31;
  const int wv   = tid >> 5;
  const int col  = lane & 15;
  const int hi   = lane >> 4;
  const int row  = col;

  v2f GiA[4], GiB[4];
#pragma unroll
  for (int kc = 0; kc < 4; ++kc) {
    const int k = kc * 4 + 2 * hi;
    GiA[kc].x = gi[row * 16 + k];
    GiA[kc].y = gi[row * 16 + k + 1];
    GiB[kc].x = gi[k * 16 + col];
    GiB[kc].y = gi[(k + 1) * 16 + col];
  }

  float* xw = xb[wv];
  float* tw = tb[wv];
  const long gw     = (long)blockIdx.x * 8 + wv;
  const long stride = (long)gridDim.x * 8;

  for (long m = gw; m < nmat; m += stride) {
    // coalesced 1KB matrix load: each lane grabs 32B
    const float4* src = (const float4*)(X + m * 256);
    const float4 p0 = src[lane * 2];
    const float4 p1 = src[lane * 2 + 1];
    wsync();                               // order vs previous iteration's LDS reads
    ((float4*)xw)[lane * 2]     = p0;
    ((float4*)xw)[lane * 2 + 1] = p1;
    wsync();

    // T = Gi * X
    v8f acc = {};
#pragma unroll
    for (int kc = 0; kc < 4; ++kc) {
      const int k = kc * 4 + 2 * hi;
      v2f b;
      b.x = xw[k * 16 + col];
      b.y = xw[(k + 1) * 16 + col];
      acc = __builtin_amdgcn_wmma_f32_16x16x4_f32(false, GiA[kc], false, b,
                                                  (short)0, acc, false, false);
    }
#pragma unroll
    for (int v = 0; v < 8; ++v) tw[(v + 8 * hi) * 17 + col] = acc[v];
    wsync();

    // Y = T * Gi
    v8f acc2 = {};
#pragma unroll
    for (int kc = 0; kc < 4; ++kc) {
      const int k = kc * 4 + 2 * hi;
      v2f a;
      a.x = tw[row * 17 + k];
      a.y = tw[row * 17 + k + 1];
      acc2 = __builtin_amdgcn_wmma_f32_16x16x4_f32(false, a, false, GiB[kc],
                                                   (short)0, acc2, false, false);
    }
    float* dst = out + m * 256;
#pragma unroll
    for (int v = 0; v < 8; ++v)
      __builtin_nontemporal_store(acc2[v], dst + (v + 8 * hi) * 16 + col);
  }
}

// ---------------------------------------------------------------------------
extern "C" void kernel_launch(void* const* d_in, const int* in_sizes, int n_in,
                              void* d_out, int out_size, void* d_ws, size_t ws_size,
                              hipStream_t stream) {
  (void)n_in; (void)out_size; (void)ws_size;
  const float* X = (const float*)d_in[0];
  const float* R = (const float*)d_in[1];
  float* out = (float*)d_out;
  float* ws  = (float*)d_ws;

  const int nmat = in_sizes[0] / 256;        // 102400

  float* part  = ws;                         // 32*16*256 = 131072 floats
  float* xmean = ws + 131072;                // 32*256   =   8192 floats
  float* gim   = ws + 139264;                // 256 floats

  hipLaunchKernelGGL(k_reduce, dim3(512), dim3(256), 0, stream, X, part);
  hipLaunchKernelGGL(k_math, dim3(1), dim3(256), 0, stream,
                     part, R, xmean, gim, out + (long)nmat * 256);
  hipLaunchKernelGGL(k_apply, dim3(800), dim3(256), 0, stream, X, gim, out, nmat);
}